// ConfigurableGATEncoder_46497315947117
// MI455X (gfx1250) — compile-verified
//
#include <hip/hip_runtime.h>
#include <hip/hip_bf16.h>

typedef __attribute__((ext_vector_type(2))) float v2f;
typedef __attribute__((ext_vector_type(8))) float v8f;

// ---------------------------------------------------------------------------
// Build x = concat(x_base[N,128], cell_emb[cell_ids][N,8]) -> [N,136]
// ---------------------------------------------------------------------------
__global__ void build_x_kernel(const float* __restrict__ xb,
                               const int* __restrict__ cid,
                               const float* __restrict__ emb,
                               float* __restrict__ x, long long total) {
    long long i = (long long)blockIdx.x * blockDim.x + threadIdx.x;
    if (i >= total) return;
    int row = (int)(i / 136);
    int col = (int)(i % 136);
    x[i] = (col < 128) ? xb[(size_t)row * 128 + col]
                       : emb[(size_t)cid[row] * 8 + (col - 128)];
}

// ---------------------------------------------------------------------------
// WMMA GEMM: C[M,128] = A[M,K] * B[K,128], fp32 via V_WMMA_F32_16X16X4_F32.
// Block = 256 threads = 8 waves; block owns 16 rows, wave w owns cols 16w..16w+15.
// A tile (16 x K) staged in LDS, reused by all 8 waves.
// ISA 7.12.2 layouts:
//   A 16x4 f32 : lanes 0-15 M=lane, v0=K0 v1=K1; lanes 16-31 M=lane-16, v0=K2 v1=K3
//   B 4x16 f32 : v0 lanes0-15=row K0, lanes16-31=row K2; v1 rows K1/K3
//   C/D        : vgpr g, lanes 0-15 -> M=g, lanes 16-31 -> M=g+8; N=lane&15
// ---------------------------------------------------------------------------
template <int K>
__global__ void gemm_wmma_kernel(const float* __restrict__ A,
                                 const float* __restrict__ B,
                                 float* __restrict__ C, int M) {
    __shared__ float As[16 * K];
    const int wave = threadIdx.x >> 5;
    const int lane = threadIdx.x & 31;
    const int half = lane >> 4;     // 0: lanes 0-15, 1: lanes 16-31
    const int l16  = lane & 15;
    const size_t mt = (size_t)blockIdx.x * 16;

    // cooperative A-tile load (16 rows, row-major contiguous)
    for (int t = threadIdx.x; t < 16 * K; t += blockDim.x) {
        int r = t / K;
        As[t] = ((long long)(mt + r) < M) ? A[mt * K + t] : 0.0f;
    }
    __syncthreads();

    const int nt = wave * 16;
    v8f acc = {};
#pragma unroll
    for (int k0 = 0; k0 < K; k0 += 4) {
        v2f a, b;
        a.x = As[l16 * K + k0 + 2 * half];
        a.y = As[l16 * K + k0 + 2 * half + 1];
        b.x = B[(size_t)(k0 + 2 * half) * 128 + nt + l16];
        b.y = B[(size_t)(k0 + 2 * half + 1) * 128 + nt + l16];
        acc = __builtin_amdgcn_wmma_f32_16x16x4_f32(
            /*neg_a=*/false, a, /*neg_b=*/false, b,
            /*c_mod=*/(short)0, acc, /*reuse_a=*/false, /*reuse_b=*/false);
    }
#pragma unroll
    for (int g = 0; g < 8; ++g) {
        size_t m = mt + g + 8 * half;
        if ((long long)m < M) C[m * 128 + nt + l16] = acc[g];
    }
}

// ---------------------------------------------------------------------------
// alpha_src[n] = h[n,:].a1 ; alpha_dst[n] = h[n,:].a2  (one wave per row)
// ---------------------------------------------------------------------------
__global__ void rowdot2_kernel(const float* __restrict__ h,
                               const float* __restrict__ a1,
                               const float* __restrict__ a2,
                               float* __restrict__ o1, float* __restrict__ o2,
                               int n) {
    int wave = threadIdx.x >> 5, lane = threadIdx.x & 31;
    int row = blockIdx.x * 8 + wave;
    if (row >= n) return;
    const float* hr = h + (size_t)row * 128;
    float s1 = 0.f, s2 = 0.f;
#pragma unroll
    for (int j = 0; j < 4; ++j) {
        float v = hr[lane + 32 * j];
        s1 += v * a1[lane + 32 * j];
        s2 += v * a2[lane + 32 * j];
    }
#pragma unroll
    for (int off = 16; off; off >>= 1) {
        s1 += __shfl_xor(s1, off, 32);
        s2 += __shfl_xor(s2, off, 32);
    }
    if (lane == 0) { o1[row] = s1; o2[row] = s2; }
}

// ---------------------------------------------------------------------------
// per-node init: segment-max (order-preserving uint encoding of -inf) + denom=0
// ---------------------------------------------------------------------------
__device__ __forceinline__ unsigned enc_f32(float f) {
    unsigned u = __float_as_uint(f);
    return (u & 0x80000000u) ? ~u : (u | 0x80000000u);
}
__device__ __forceinline__ float dec_f32(unsigned u) {
    return (u & 0x80000000u) ? __uint_as_float(u ^ 0x80000000u)
                             : __uint_as_float(~u);
}

__global__ void init_nodes_kernel(unsigned* __restrict__ menc,
                                  float* __restrict__ denom, int n) {
    int i = blockIdx.x * blockDim.x + threadIdx.x;
    if (i >= n) return;
    menc[i] = 0x007FFFFFu;   // enc(-inf)
    denom[i] = 0.0f;
}

__global__ void zero_f32_kernel(float* __restrict__ p, long long total) {
    long long i = (long long)blockIdx.x * blockDim.x + threadIdx.x;
    if (i < total) p[i] = 0.0f;
}

// ---------------------------------------------------------------------------
// pass 1: e = leaky_relu(aS[src]+aD[dst]); atomicMax of encoded e per dst
// ---------------------------------------------------------------------------
__global__ void edge_score_kernel(const int* __restrict__ ei, int E, int Etot,
                                  const float* __restrict__ aS,
                                  const float* __restrict__ aD,
                                  float* __restrict__ ebuf,
                                  unsigned* __restrict__ menc) {
    int idx = blockIdx.x * blockDim.x + threadIdx.x;
    if (idx >= Etot) return;
    int s, d;
    if (idx < E) { s = ei[idx]; d = ei[E + idx]; } else { s = d = idx - E; }
    float e = aS[s] + aD[d];
    e = e > 0.0f ? e : 0.2f * e;
    ebuf[idx] = e;
    atomicMax(menc + d, enc_f32(e));
}

// ---------------------------------------------------------------------------
// pass 2: ex = exp(e - m[dst]) (in place); atomicAdd into denom[dst]
// ---------------------------------------------------------------------------
__global__ void edge_exp_kernel(const int* __restrict__ ei, int E, int Etot,
                                float* __restrict__ ebuf,
                                const unsigned* __restrict__ menc,
                                float* __restrict__ denom) {
    int idx = blockIdx.x * blockDim.x + threadIdx.x;
    if (idx >= Etot) return;
    int d = (idx < E) ? ei[E + idx] : idx - E;
    float ex = __expf(ebuf[idx] - dec_f32(menc[d]));
    ebuf[idx] = ex;
    atomicAdd(denom + d, ex);
}

// ---------------------------------------------------------------------------
// pass 3: out[dst,:] += h[src,:] * (ex/denom[dst]); one wave per edge,
// float4 gather per lane -> 4 coalesced f32 atomics (L2-resident working set)
// ---------------------------------------------------------------------------
__global__ void edge_agg_kernel(const int* __restrict__ ei, int E, int Etot,
                                const float* __restrict__ ebuf,
                                const float* __restrict__ denom,
                                const float* __restrict__ h,
                                float* __restrict__ out) {
    long long gid = (long long)blockIdx.x * blockDim.x + threadIdx.x;
    int edge = (int)(gid >> 5);
    int lane = (int)(gid & 31);
    if (edge >= Etot) return;
    int s, d;
    if (edge < E) { s = ei[edge]; d = ei[E + edge]; } else { s = d = edge - E; }
    float alpha = ebuf[edge] / denom[d];
    const float4 hv = *(const float4*)(h + (size_t)s * 128 + lane * 4);
    float* op = out + (size_t)d * 128 + lane * 4;
    atomicAdd(op + 0, hv.x * alpha);
    atomicAdd(op + 1, hv.y * alpha);
    atomicAdd(op + 2, hv.z * alpha);
    atomicAdd(op + 3, hv.w * alpha);
}

// ---------------------------------------------------------------------------
// x2 = ELU(LayerNorm(acc + b1))  — one wave per row, 4 elems/lane
// ---------------------------------------------------------------------------
__global__ void bias_ln_elu_kernel(const float* __restrict__ acc,
                                   const float* __restrict__ b,
                                   const float* __restrict__ g,
                                   const float* __restrict__ beta,
                                   float* __restrict__ x2, int n) {
    int wave = threadIdx.x >> 5, lane = threadIdx.x & 31;
    int row = blockIdx.x * 8 + wave;
    if (row >= n) return;
    const float* r = acc + (size_t)row * 128;
    float v[4];
    float sum = 0.f;
#pragma unroll
    for (int i = 0; i < 4; ++i) {
        v[i] = r[lane * 4 + i] + b[lane * 4 + i];
        sum += v[i];
    }
#pragma unroll
    for (int off = 16; off; off >>= 1) sum += __shfl_xor(sum, off, 32);
    float mu = sum * (1.0f / 128.0f);
    float vs = 0.f;
#pragma unroll
    for (int i = 0; i < 4; ++i) { float dlt = v[i] - mu; vs += dlt * dlt; }
#pragma unroll
    for (int off = 16; off; off >>= 1) vs += __shfl_xor(vs, off, 32);
    float rstd = rsqrtf(vs * (1.0f / 128.0f) + 1e-5f);
#pragma unroll
    for (int i = 0; i < 4; ++i) {
        float y = (v[i] - mu) * rstd * g[lane * 4 + i] + beta[lane * 4 + i];
        x2[(size_t)row * 128 + lane * 4 + i] = y > 0.0f ? y : __expf(y) - 1.0f;
    }
}

// ---------------------------------------------------------------------------
// d_out += b2 (broadcast over rows)
// ---------------------------------------------------------------------------
__global__ void add_bias_kernel(float* __restrict__ out,
                                const float* __restrict__ b, long long total) {
    long long i = (long long)blockIdx.x * blockDim.x + threadIdx.x;
    if (i < total) out[i] += b[i & 127];
}

// ---------------------------------------------------------------------------
extern "C" void kernel_launch(void* const* d_in, const int* in_sizes, int n_in,
                              void* d_out, int out_size, void* d_ws, size_t ws_size,
                              hipStream_t stream) {
    const float* xb  = (const float*)d_in[0];
    const int*   cid = (const int*)d_in[1];
    const int*   ei  = (const int*)d_in[2];
    const float* emb = (const float*)d_in[3];
    const float* W1  = (const float*)d_in[4];
    const float* as1 = (const float*)d_in[5];
    const float* ad1 = (const float*)d_in[6];
    const float* b1  = (const float*)d_in[7];
    const float* lng = (const float*)d_in[8];
    const float* lnb = (const float*)d_in[9];
    const float* W2  = (const float*)d_in[10];
    const float* as2 = (const float*)d_in[11];
    const float* ad2 = (const float*)d_in[12];
    const float* b2  = (const float*)d_in[13];
    float* out = (float*)d_out;

    const int N = in_sizes[0] / 128;
    const int E = in_sizes[2] / 2;
    const int Etot = E + N;

    // ---- workspace carve-out (all L2-resident; ~108 MB total) ----
    char* ws = (char*)d_ws;
    size_t off = 0;
    auto carve = [&](size_t bytes) {
        char* p = ws + off;
        off += (bytes + 255) & ~(size_t)255;
        return p;
    };
    float*    x136  = (float*)carve((size_t)N * 136 * 4);
    float*    h     = (float*)carve((size_t)N * 128 * 4);
    float*    x2    = (float*)carve((size_t)N * 128 * 4);
    float*    acc1  = (float*)carve((size_t)N * 128 * 4);
    float*    aS    = (float*)carve((size_t)N * 4);
    float*    aD    = (float*)carve((size_t)N * 4);
    unsigned* menc  = (unsigned*)carve((size_t)N * 4);
    float*    denom = (float*)carve((size_t)N * 4);
    float*    ebuf  = (float*)carve((size_t)Etot * 4);

    const int TPB = 256;
    const long long nx  = (long long)N * 136;
    const long long nc  = (long long)N * 128;
    const int gemm_blocks = (N + 15) / 16;
    const int row_blocks  = (N + 7) / 8;
    const int node_blocks = (N + TPB - 1) / TPB;
    const int feat_blocks = (int)((nc + TPB - 1) / TPB);
    const int edge_blocks = (Etot + TPB - 1) / TPB;
    const int agg_blocks  = (int)(((long long)Etot * 32 + TPB - 1) / TPB);

    // ---------------- layer 1 ----------------
    build_x_kernel<<<(int)((nx + TPB - 1) / TPB), TPB, 0, stream>>>(xb, cid, emb, x136, nx);
    gemm_wmma_kernel<136><<<gemm_blocks, TPB, 0, stream>>>(x136, W1, h, N);
    rowdot2_kernel<<<row_blocks, TPB, 0, stream>>>(h, as1, ad1, aS, aD, N);
    init_nodes_kernel<<<node_blocks, TPB, 0, stream>>>(menc, denom, N);
    zero_f32_kernel<<<feat_blocks, TPB, 0, stream>>>(acc1, nc);
    edge_score_kernel<<<edge_blocks, TPB, 0, stream>>>(ei, E, Etot, aS, aD, ebuf, menc);
    edge_exp_kernel<<<edge_blocks, TPB, 0, stream>>>(ei, E, Etot, ebuf, menc, denom);
    edge_agg_kernel<<<agg_blocks, TPB, 0, stream>>>(ei, E, Etot, ebuf, denom, h, acc1);
    bias_ln_elu_kernel<<<row_blocks, TPB, 0, stream>>>(acc1, b1, lng, lnb, x2, N);

    // ---------------- layer 2 ----------------
    gemm_wmma_kernel<128><<<gemm_blocks, TPB, 0, stream>>>(x2, W2, h, N);
    rowdot2_kernel<<<row_blocks, TPB, 0, stream>>>(h, as2, ad2, aS, aD, N);
    init_nodes_kernel<<<node_blocks, TPB, 0, stream>>>(menc, denom, N);
    zero_f32_kernel<<<feat_blocks, TPB, 0, stream>>>(out, nc);
    edge_score_kernel<<<edge_blocks, TPB, 0, stream>>>(ei, E, Etot, aS, aD, ebuf, menc);
    edge_exp_kernel<<<edge_blocks, TPB, 0, stream>>>(ei, E, Etot, ebuf, menc, denom);
    edge_agg_kernel<<<agg_blocks, TPB, 0, stream>>>(ei, E, Etot, ebuf, denom, h, out);
    add_bias_kernel<<<feat_blocks, TPB, 0, stream>>>(out, b2, nc);
}